// GptOssTopKRouter_71459665871174
// MI455X (gfx1250) — compile-verified
//
#include <hip/hip_runtime.h>

// ---------------------------------------------------------------------------
// GptOssTopKRouter for MI455X (gfx1250, wave32, WMMA + async global->LDS)
// logits = H[16384x2880] * W^T[2880x128] + bias ; per-token top-4 softmax
// Fused: GEMM (V_WMMA_F32_16X16X4_F32, fp32-exact) -> top-k -> softmax ->
// scatter. W chunks are double-buffered in LDS via GLOBAL_LOAD_ASYNC_TO_LDS
// (ASYNCcnt), overlapping the L2 fill of chunk c+1 with WMMA on chunk c.
// ---------------------------------------------------------------------------

typedef __attribute__((ext_vector_type(2))) float v2f;
typedef __attribute__((ext_vector_type(8))) float v8f;

#define NUM_TOKENS   16384
#define NUM_EXPERTS  128
#define HIDDEN       2880
#define TOPK         4

#define KC           48            // K-chunk staged in LDS (2880 = 60 * 48)
#define NCHUNK       (HIDDEN / KC) // 60
#define WPAD         52            // padded LDS row stride (dwords): row*52 mod 64
                                   // spans all multiples of 4 -> conflict-free f2 reads
#define LPAD         133           // padded LDS row stride (dwords) for logits
#define THREADS      128           // 4 wave32s
#define TOK_PER_WAVE 16
#define TOK_PER_WG   64            // 4 waves * 16 tokens
#define NT           8             // 128 experts / 16-wide N tiles

#define BUFFLOATS    (NUM_EXPERTS * WPAD)      // 6656 floats per buffer
#define BUFBYTES     (BUFFLOATS * 4)           // 26624 B
#define F4_PER_CHUNK (NUM_EXPERTS * (KC / 4))  // 1536 float4 per chunk
#define F4_PER_THR   (F4_PER_CHUNK / THREADS)  // 12 async loads / thread / chunk

__global__ __launch_bounds__(THREADS)
void router_topk_kernel(const float* __restrict__ H,
                        const float* __restrict__ W,
                        const float* __restrict__ bias,
                        float* __restrict__ out) {
  __shared__ float smem[2 * BUFFLOATS];        // 53248 B; logits region reuses it

  const int tid  = threadIdx.x;
  const int lane = tid & 31;
  const int wv   = tid >> 5;       // wave id 0..3
  const int n    = lane & 15;      // column-in-tile / row-in-A-tile
  const int hi   = lane >> 4;      // half-wave select (K pair)
  const long tokBase = (long)blockIdx.x * TOK_PER_WG;

  // Low 32 bits of a flat shared pointer are the LDS byte offset (aperture
  // keeps the LDS base purely in addr[63:32]).
  const unsigned smemBase = (unsigned)(size_t)smem;

  // Per-thread async-load assignment for one W chunk (row-major, float4).
  unsigned ldsOff[F4_PER_THR];
  long     gOff[F4_PER_THR];
#pragma unroll
  for (int i = 0; i < F4_PER_THR; ++i) {
    const int idx = i * THREADS + tid;
    const int row = idx / (KC / 4);
    const int c4  = idx - row * (KC / 4);
    ldsOff[i] = (unsigned)((row * WPAD + 4 * c4) * 4);
    gOff[i]   = (long)row * HIDDEN + 4 * c4;
  }

  // 8 accumulator tiles: tokens [wv*16 .. wv*16+15] x experts [16t .. 16t+15]
  v8f acc[NT] = {};

  // A fragment source: lane holds H[token = n][K = k + 2*hi + {0,1}]
  const float* aptr = H + (tokBase + (long)wv * TOK_PER_WAVE + n) * (long)HIDDEN + 2 * hi;
  // B fragment base inside an LDS W-chunk buffer.
  const int bbase = n * WPAD + 2 * hi;

  // ---- prologue: async-stage chunk 0 into buffer 0 ------------------------
#pragma unroll
  for (int i = 0; i < F4_PER_THR; ++i) {
    const unsigned long long src = (unsigned long long)(size_t)(W + gOff[i]);
    const unsigned dst = smemBase + ldsOff[i];
    asm volatile("global_load_async_to_lds_b128 %0, %1, off"
                 :: "v"(dst), "v"(src) : "memory");
  }
  asm volatile("s_wait_asynccnt 0x0" ::: "memory");
  __syncthreads();

  unsigned buf = 0;
  for (int c = 0; c < NCHUNK; ++c) {
    // Issue next chunk into the other buffer (overlaps with WMMA below).
    if (c + 1 < NCHUNK) {
      const long kn = (long)(c + 1) * KC;
      const unsigned dbase = smemBase + (buf ^ 1) * BUFBYTES;
#pragma unroll
      for (int i = 0; i < F4_PER_THR; ++i) {
        const unsigned long long src =
            (unsigned long long)(size_t)(W + kn + gOff[i]);
        const unsigned dst = dbase + ldsOff[i];
        asm volatile("global_load_async_to_lds_b128 %0, %1, off"
                     :: "v"(dst), "v"(src) : "memory");
      }
    }

    // WMMA on current buffer.
    const float* bp0 = smem + buf * BUFFLOATS + bbase;
    const float* ap  = aptr + (long)c * KC;
#pragma unroll 4
    for (int kk = 0; kk < KC; kk += 4) {
      const v2f a = *(const v2f*)(ap + kk);
#pragma unroll
      for (int t = 0; t < NT; ++t) {
        const v2f b = *(const v2f*)(bp0 + kk + t * (16 * WPAD));
        acc[t] = __builtin_amdgcn_wmma_f32_16x16x4_f32(
            /*neg_a=*/false, a, /*neg_b=*/false, b,
            /*c_mod=*/(short)0, acc[t], /*reuse_a=*/false, /*reuse_b=*/false);
      }
    }

    asm volatile("s_wait_asynccnt 0x0" ::: "memory");
    __syncthreads();   // next chunk resident; everyone done reading current
    buf ^= 1;
  }

  // Spill C tiles to LDS as logits[token_local][expert].
  // C layout: VGPR p -> token row (p + 8*hi), lane n -> expert column.
#pragma unroll
  for (int t = 0; t < NT; ++t) {
#pragma unroll
    for (int p = 0; p < 8; ++p) {
      const int tokLocal = wv * TOK_PER_WAVE + p + 8 * hi;
      smem[tokLocal * LPAD + t * 16 + n] = acc[t][p];
    }
  }
  __syncthreads();

  float* oScores = out;
  float* oIdx    = out + (long)NUM_TOKENS * NUM_EXPERTS;

  if (tid < TOK_PER_WG) {
    float* lrow = smem + tid * LPAD;

    float tv[TOPK];
    int   ti[TOPK];
#pragma unroll
    for (int j = 0; j < TOPK; ++j) { tv[j] = -3.402823466e38f; ti[j] = 0; }

    // Strict-> insertion keeps lower index first on ties (matches lax.top_k).
    for (int e = 0; e < NUM_EXPERTS; ++e) {
      const float v = lrow[e] + bias[e];
      if (v > tv[TOPK - 1]) {
        tv[TOPK - 1] = v; ti[TOPK - 1] = e;
#pragma unroll
        for (int j = TOPK - 1; j > 0; --j) {
          if (tv[j] > tv[j - 1]) {
            const float fv = tv[j]; tv[j] = tv[j - 1]; tv[j - 1] = fv;
            const int   iv = ti[j]; ti[j] = ti[j - 1]; ti[j - 1] = iv;
          }
        }
      }
    }

    // Softmax over the 4 kept logits.
    const float m = tv[0];
    float ex[TOPK];
    float s = 0.0f;
#pragma unroll
    for (int j = 0; j < TOPK; ++j) { ex[j] = expf(tv[j] - m); s += ex[j]; }
    const float inv = 1.0f / s;

    // Rewrite this token's LDS row as the sparse score row.
    for (int e = 0; e < NUM_EXPERTS; ++e) lrow[e] = 0.0f;
#pragma unroll
    for (int j = 0; j < TOPK; ++j) lrow[ti[j]] = ex[j] * inv;

    // Indices (tuple output concatenated after scores; promoted to f32).
    const long ib = (tokBase + tid) * TOPK;
#pragma unroll
    for (int j = 0; j < TOPK; ++j) oIdx[ib + j] = (float)ti[j];
  }
  __syncthreads();

  // Coalesced store of the block's 64x128 score rows.
  for (int i = tid; i < TOK_PER_WG * NUM_EXPERTS; i += THREADS) {
    const int r = i >> 7;
    const int c = i & 127;
    oScores[tokBase * NUM_EXPERTS + i] = smem[r * LPAD + c];
  }
}

extern "C" void kernel_launch(void* const* d_in, const int* in_sizes, int n_in,
                              void* d_out, int out_size, void* d_ws, size_t ws_size,
                              hipStream_t stream) {
  const float* H    = (const float*)d_in[0];  // [16384, 2880]
  const float* W    = (const float*)d_in[1];  // [128, 2880]
  const float* bias = (const float*)d_in[2];  // [128]
  float* out = (float*)d_out;                 // [16384*128] scores ++ [16384*4] idx

  const int grid = NUM_TOKENS / TOK_PER_WG;   // 256 blocks, 4 waves each
  router_topk_kernel<<<grid, THREADS, 0, stream>>>(H, W, bias, out);
}